// DisentangledSeqEncoder_3118146257580
// MI455X (gfx1250) — compile-verified
//
#include <hip/hip_runtime.h>

typedef float v2f __attribute__((ext_vector_type(2)));
typedef float v8f __attribute__((ext_vector_type(8)));

#define B_  512
#define T_  1024
#define D_  64
#define K_  16
#define NTHREADS 512
#define EPS_ 1e-6f
#define SCALE_ 0.125f   // 1/sqrt(64)

// ---- LDS layout (bytes) ----
constexpr int OFF_Z   = 0;                       // 1024*64 f32   = 262144
constexpr int OFF_E   = OFF_Z   + T_*D_*4;       // 1024*16 f16   =  32768
constexpr int OFF_MU  = OFF_E   + T_*K_*2;       // 1024 f32
constexpr int OFF_RS  = OFF_MU  + T_*4;          // 1024 f32
constexpr int OFF_S   = OFF_RS  + T_*4;          // 1024 f32 (scores_t -> p_i)
constexpr int OFF_PNP = OFF_S   + T_*4;          // 16*64 f32 (pn * g0)
constexpr int OFF_P1  = OFF_PNP + K_*D_*4;       // 16 f32
constexpr int OFF_BD  = OFF_P1  + K_*4;          // 16 f32
constexpr int OFF_QH  = OFF_BD  + K_*4;          // 64 f32 (qhat)
constexpr int OFF_G2  = OFF_QH  + D_*4;          // 64 f32
constexpr int OFF_B2  = OFF_G2  + D_*4;
constexpr int OFF_G4  = OFF_B2  + D_*4;
constexpr int OFF_B4  = OFF_G4  + D_*4;
constexpr int OFF_ACC = OFF_B4  + D_*4;          // 16*64 f32 out accumulator
constexpr int OFF_RED = OFF_ACC + K_*D_*4;       // 64 f32 reduction scratch
constexpr int SMEM_BYTES = OFF_RED + 64*4;       // ~310 KB < 320 KB/WGP

__device__ __forceinline__ float wred_add(float v) {
  #pragma unroll
  for (int m = 16; m >= 1; m >>= 1) v += __shfl_xor(v, m, 32);
  return v;
}

__global__ __launch_bounds__(NTHREADS, 1)
void disentangled_encoder_kernel(const float* __restrict__ zg,
                                 const float* __restrict__ protos,
                                 const float* __restrict__ alphas,
                                 const float* __restrict__ b_bias,
                                 const float* __restrict__ W,
                                 const float* __restrict__ Wb,
                                 const float* __restrict__ ln_g,
                                 const float* __restrict__ ln_b,
                                 const float* __restrict__ beta_seq,
                                 float* __restrict__ out)
{
  extern __shared__ char smem[];
  float*    sZ   = (float*)(smem + OFF_Z);
  _Float16* sE   = (_Float16*)(smem + OFF_E);
  float*    sMu  = (float*)(smem + OFF_MU);
  float*    sRs  = (float*)(smem + OFF_RS);
  float*    sS   = (float*)(smem + OFF_S);
  float*    sPnp = (float*)(smem + OFF_PNP);
  float*    sP1  = (float*)(smem + OFF_P1);
  float*    sBd  = (float*)(smem + OFF_BD);
  float*    sQh  = (float*)(smem + OFF_QH);
  float*    sG2  = (float*)(smem + OFF_G2);
  float*    sB2  = (float*)(smem + OFF_B2);
  float*    sG4  = (float*)(smem + OFF_G4);
  float*    sB4  = (float*)(smem + OFF_B4);
  float*    sAcc = (float*)(smem + OFF_ACC);
  float*    sRed = (float*)(smem + OFF_RED);

  const int b    = blockIdx.x;
  const int tid  = threadIdx.x;
  const int wave = tid >> 5;
  const int lane = tid & 31;

  // ---------------- Phase 0: async-DMA z[b] into LDS (CDNA5 async pipe) ----
  // GLOBAL_LOAD_ASYNC_TO_LDS_B128: no VGPR data movement, tracked by ASYNCcnt.
  {
    const float4* zsrc = (const float4*)(zg + (size_t)b * T_ * D_);
    #pragma unroll 4
    for (int i = tid; i < T_*D_/4; i += NTHREADS) {
      unsigned lds_addr = (unsigned)(uintptr_t)(sZ + 4*i);      // LDS offset = flat[31:0]
      unsigned long long gaddr = (unsigned long long)(uintptr_t)(zsrc + i);
      asm volatile("global_load_async_to_lds_b128 %0, %1, off"
                   :: "v"(lds_addr), "v"(gaddr) : "memory");
    }
    // overlap: zero the out accumulator + fetch LN params while DMA is in flight
    for (int i = tid; i < K_*D_;  i += NTHREADS) sAcc[i] = 0.f;
    if (tid < D_) {
      sG2[tid] = ln_g[2*D_+tid]; sB2[tid] = ln_b[2*D_+tid];
      sG4[tid] = ln_g[4*D_+tid]; sB4[tid] = ln_b[4*D_+tid];
    }
    asm volatile("s_wait_asynccnt 0x0" ::: "memory");   // my async loads done
  }
  __syncthreads();                                      // all waves' loads visible

  // ---------------- Setup: prototypes LN (folded with g0/b0), row stats, q/qhat
  {
    // wave w handles prototype row k=w (16 waves, K=16)
    const int k = wave;
    float x1 = protos[k*D_+lane], x2 = protos[k*D_+lane+32];
    float s1 = wred_add(x1 + x2);
    float s2 = wred_add(x1*x1 + x2*x2);
    float mu = s1 * (1.f/D_);
    float rs = rsqrtf(s2*(1.f/D_) - mu*mu + EPS_);
    float pn1 = (x1-mu)*rs*ln_g[D_+lane]    + ln_b[D_+lane];
    float pn2 = (x2-mu)*rs*ln_g[D_+lane+32] + ln_b[D_+lane+32];
    float pp1 = pn1 * ln_g[lane];       // fold gamma0 into pn
    float pp2 = pn2 * ln_g[lane+32];
    sPnp[k*D_+lane] = pp1; sPnp[k*D_+lane+32] = pp2;
    float p1 = wred_add(pp1 + pp2);                         // sum_d g0*pn
    float bd = wred_add(ln_b[lane]*pn1 + ln_b[lane+32]*pn2);// sum_d b0*pn
    if (lane == 0) { sP1[k] = p1; sBd[k] = bd; }
  }
  // per-row mean/rstd of z (feeds the rank-1 LN epilogue of the scores GEMM)
  for (int t = wave*64; t < wave*64 + 64; ++t) {
    float x1 = sZ[t*D_+lane], x2 = sZ[t*D_+lane+32];
    float s1 = wred_add(x1 + x2);
    float s2 = wred_add(x1*x1 + x2*x2);
    if (lane == 0) {
      float mu = s1 * (1.f/D_);
      sMu[t] = mu;
      sRs[t] = rsqrtf(s2*(1.f/D_) - mu*mu + EPS_);
    }
  }
  if (wave == 0) {
    // q = LN(alphas[-1] + z[:, -1] + b_bias);  qhat = q + W q;  c0 = Wb . q
    float x1 = alphas[1023*D_+lane]    + sZ[1023*D_+lane]    + b_bias[lane];
    float x2 = alphas[1023*D_+lane+32] + sZ[1023*D_+lane+32] + b_bias[lane+32];
    float s1 = wred_add(x1 + x2);
    float s2 = wred_add(x1*x1 + x2*x2);
    float mu = s1 * (1.f/D_);
    float rs = rsqrtf(s2*(1.f/D_) - mu*mu + EPS_);
    float q1 = (x1-mu)*rs*ln_g[3*D_+lane]    + ln_b[3*D_+lane];
    float q2 = (x2-mu)*rs*ln_g[3*D_+lane+32] + ln_b[3*D_+lane+32];
    float a1 = 0.f, a2 = 0.f;
    #pragma unroll 4
    for (int e = 0; e < 32; ++e) {
      float qe  = __shfl(q1, e, 32);
      float qe2 = __shfl(q2, e, 32);
      a1 += W[lane*D_+e]*qe      + W[lane*D_+e+32]*qe2;
      a2 += W[(lane+32)*D_+e]*qe + W[(lane+32)*D_+e+32]*qe2;
    }
    sQh[lane]    = q1 + a1;
    sQh[lane+32] = q2 + a2;
    float c0p = wred_add(Wb[lane]*q1 + Wb[lane+32]*q2);
    if (lane == 0) sRed[32] = c0p;
  }
  __syncthreads();

  // ---------------- Phase A: scores_k = z @ pn'^T  (f32 WMMA, exact) --------
  // zn.pn folded: score = rs_t*(Zdot - mu_t*P1[k]) + bdot[k]
  {
    const int half = lane >> 4;     // K-slice selector within A/B operands
    const int l16  = lane & 15;
    v2f bb[16];
    #pragma unroll
    for (int i = 0; i < 16; ++i) {
      int d = 4*i + 2*half;
      bb[i] = *(const v2f*)&sPnp[l16*D_ + d];   // B[d][k] = pn'[k][d]
    }
    const float P1k = sP1[l16];
    const float bdk = sBd[l16];
    for (int st = 0; st < 4; ++st) {
      const int t0 = wave*64 + st*16;
      v8f c = {};
      #pragma unroll
      for (int i = 0; i < 16; ++i) {
        int d = 4*i + 2*half;
        v2f a = *(const v2f*)&sZ[(t0 + l16)*D_ + d];  // A[t][d] = raw z
        c = __builtin_amdgcn_wmma_f32_16x16x4_f32(false, a, false, bb[i],
                                                  (short)0, c, false, false);
      }
      // epilogue: LN-fold + softmax over K (lanes 0-15 / 16-31 groups)
      #pragma unroll
      for (int r = 0; r < 8; ++r) {
        int t = t0 + r + 8*half;
        float mu = sMu[t], rs = sRs[t];
        float sc = (rs*(c[r] - mu*P1k) + bdk) * SCALE_;
        float m = sc;
        #pragma unroll
        for (int msk = 8; msk >= 1; msk >>= 1) m = fmaxf(m, __shfl_xor(m, msk, 32));
        float e = __expf(sc - m);
        float S = e;
        #pragma unroll
        for (int msk = 8; msk >= 1; msk >>= 1) S += __shfl_xor(S, msk, 32);
        sE[t*K_ + l16] = (_Float16)(e / S);   // p_{k|i}
      }
    }
  }

  // ---------------- Phase A3: s_t = LN(alphas+z) . qhat + c0 (Dense folded) -
  {
    const float c0 = sRed[32];
    for (int t = wave*64; t < wave*64 + 64; ++t) {
      float z1 = sZ[t*D_+lane], z2 = sZ[t*D_+lane+32];
      float x1 = alphas[t*D_+lane]    + z1;     // alphas is L2-resident
      float x2 = alphas[t*D_+lane+32] + z2;
      float s1 = wred_add(x1 + x2);
      float s2 = wred_add(x1*x1 + x2*x2);
      float mu = s1 * (1.f/D_);
      float rs = rsqrtf(s2*(1.f/D_) - mu*mu + EPS_);
      float k1 = (x1-mu)*rs*sG2[lane]    + sB2[lane];
      float k2 = (x2-mu)*rs*sG2[lane+32] + sB2[lane+32];
      float dp = wred_add(k1*sQh[lane] + k2*sQh[lane+32]);
      if (lane == 0) sS[t] = (dp + c0) * SCALE_;
    }
  }
  __syncthreads();

  // ---------------- softmax over T (block reduction) ------------------------
  {
    float sa = sS[tid], sb = sS[tid + NTHREADS];
    float m = fmaxf(sa, sb);
    #pragma unroll
    for (int msk = 16; msk >= 1; msk >>= 1) m = fmaxf(m, __shfl_xor(m, msk, 32));
    if (lane == 0) sRed[wave] = m;
    __syncthreads();
    float M = -1e30f;
    #pragma unroll
    for (int i = 0; i < 16; ++i) M = fmaxf(M, sRed[i]);
    float ea = __expf(sa - M), eb = __expf(sb - M);
    float S = wred_add(ea + eb);
    if (lane == 0) sRed[16 + wave] = S;
    __syncthreads();
    float Ssum = 0.f;
    #pragma unroll
    for (int i = 0; i < 16; ++i) Ssum += sRed[16 + i];
    float inv = 1.f / Ssum;
    sS[tid]            = ea * inv;   // p_i
    sS[tid + NTHREADS] = eb * inv;
  }
  __syncthreads();

  // ---------------- Phase B: out[k,d] = sum_t (p_{k|i}*p_i) * z  (f32 WMMA) -
  {
    const int half = lane >> 4;
    const int l16  = lane & 15;
    const int tw   = wave*64;           // each wave owns a 64-wide T slice
    v8f acc[4] = {};
    for (int stp = 0; stp < 16; ++stp) {
      int tA = tw + 4*stp + 2*half;
      float p0 = sS[tA], p1 = sS[tA+1];
      v2f a;
      a[0] = (float)sE[tA*K_ + l16]     * p0;   // A[k][t] = w[t][k]
      a[1] = (float)sE[(tA+1)*K_ + l16] * p1;
      #pragma unroll
      for (int dt = 0; dt < 4; ++dt) {
        int d = dt*16 + l16;
        v2f bv;
        bv[0] = sZ[tA*D_ + d];                  // B[t][d] = z
        bv[1] = sZ[(tA+1)*D_ + d];
        acc[dt] = __builtin_amdgcn_wmma_f32_16x16x4_f32(false, a, false, bv,
                                                        (short)0, acc[dt], false, false);
      }
    }
    // cross-wave K-dim reduction into sAcc
    #pragma unroll
    for (int dt = 0; dt < 4; ++dt) {
      #pragma unroll
      for (int r = 0; r < 8; ++r) {
        int k = r + 8*half;
        int d = dt*16 + l16;
        atomicAdd(&sAcc[k*D_ + d], acc[dt][r]);
      }
    }
  }
  __syncthreads();

  // ---------------- Phase C: final LN(beta_seq + out) + store ---------------
  {
    const int k = wave;                 // 16 waves, 16 rows
    float v1 = beta_seq[k*D_+lane]    + sAcc[k*D_+lane];
    float v2 = beta_seq[k*D_+lane+32] + sAcc[k*D_+lane+32];
    float s1 = wred_add(v1 + v2);
    float s2 = wred_add(v1*v1 + v2*v2);
    float mu = s1 * (1.f/D_);
    float rs = rsqrtf(s2*(1.f/D_) - mu*mu + EPS_);
    float y1 = (v1-mu)*rs*sG4[lane]    + sB4[lane];
    float y2 = (v2-mu)*rs*sG4[lane+32] + sB4[lane+32];
    float* o = out + (size_t)b*K_*D_ + k*D_;
    o[lane]    = y1;
    o[lane+32] = y2;
  }
}

extern "C" void kernel_launch(void* const* d_in, const int* in_sizes, int n_in,
                              void* d_out, int out_size, void* d_ws, size_t ws_size,
                              hipStream_t stream) {
  const float* z        = (const float*)d_in[0];
  const float* protos   = (const float*)d_in[1];
  const float* alphas   = (const float*)d_in[2];
  const float* b_bias   = (const float*)d_in[3];
  const float* W        = (const float*)d_in[4];
  const float* Wb       = (const float*)d_in[5];
  const float* ln_g     = (const float*)d_in[6];
  const float* ln_b     = (const float*)d_in[7];
  const float* beta_seq = (const float*)d_in[8];
  float* out = (float*)d_out;

  disentangled_encoder_kernel<<<B_, NTHREADS, SMEM_BYTES, stream>>>(
      z, protos, alphas, b_bias, W, Wb, ln_g, ln_b, beta_seq, out);
}